// Earth_Movers_Distance_Loss_52561809769255
// MI455X (gfx1250) — compile-verified
//
#include <hip/hip_runtime.h>
#include <hip/hip_bf16.h>
#include <math.h>
#include <stdint.h>

// CDNA5 / gfx1250, wave32.
// EMD loss: out = (1/B) * sum_s sqrt( (1/C) * sum_i cdf_i^2 ),  cdf = cumsum(p-q).
// cumsum expressed as lower-triangular matmul -> V_WMMA_F32_16X16X4_F32,
// 16 samples per wave per tile, FP32 accumulate.
// Tiles staged to LDS via GLOBAL_LOAD_ASYNC_TO_LDS_B128 (ASYNCcnt) when available.

typedef __attribute__((ext_vector_type(2))) float v2f;
typedef __attribute__((ext_vector_type(8))) float v8f;

#if __has_builtin(__builtin_amdgcn_global_load_async_to_lds_b128)
#define HAVE_ASYNC_LDS 1
// Builtin signature (from hipcc diagnostic): param0 = global int4-vec*, param1 = LDS int4-vec*.
typedef int v4i_t __attribute__((vector_size(4 * sizeof(int))));
typedef __attribute__((address_space(1))) v4i_t gv4i;   // global (__device__)
typedef __attribute__((address_space(3))) v4i_t lv4i;   // LDS (__shared__)
#endif

#define C_DIM 10
#define SPB 128                         // samples per block-tile (8 waves * 16)
#define TILE_FLOATS (SPB * C_DIM)       // 1280 floats per tensor per tile
#define TILE_F4 (TILE_FLOATS / 4)       // 320 float4 chunks per tensor per tile

__global__ void emd_zero_out(float* out) {
    if (threadIdx.x == 0 && blockIdx.x == 0) out[0] = 0.0f;
}

__global__ __launch_bounds__(256)
void emd_wmma_kernel(const float* __restrict__ p, const float* __restrict__ q,
                     float* __restrict__ out, int ntiles, float invB) {
#ifdef HAVE_ASYNC_LDS
    __shared__ __align__(16) float p_tile[TILE_FLOATS];
    __shared__ __align__(16) float q_tile[TILE_FLOATS];
#else
    __shared__ __align__(16) float d_tile[TILE_FLOATS];
#endif
    __shared__ float block_acc;

    const int tid  = threadIdx.x;
    const int lane = tid & 31;
    const int wave = tid >> 5;
    const int n    = lane & 15;   // N column (sample within wave group) == A-row M
    const int hi   = lane >> 4;   // half-wave: selects K offset (+2) and C/D rows (+8)

    if (tid == 0) block_acc = 0.0f;

    for (int tile = blockIdx.x; tile < ntiles; tile += gridDim.x) {
        __syncthreads();  // LDS tile reuse barrier (also orders block_acc init)

        const size_t base = (size_t)tile * TILE_FLOATS;
        const float4* p4 = (const float4*)(p + base);
        const float4* q4 = (const float4*)(q + base);

#ifdef HAVE_ASYNC_LDS
        // ---- async DMA: global -> LDS, no VGPR round-trip (ASYNCcnt) ----
        {
            float4* lp = (float4*)p_tile;
            float4* lq = (float4*)q_tile;
            __builtin_amdgcn_global_load_async_to_lds_b128(
                (gv4i*)(uintptr_t)(p4 + tid),
                (lv4i*)(unsigned)(uintptr_t)(lp + tid), 0, 0);
            __builtin_amdgcn_global_load_async_to_lds_b128(
                (gv4i*)(uintptr_t)(q4 + tid),
                (lv4i*)(unsigned)(uintptr_t)(lq + tid), 0, 0);
            if (tid < TILE_F4 - 256) {
                __builtin_amdgcn_global_load_async_to_lds_b128(
                    (gv4i*)(uintptr_t)(p4 + tid + 256),
                    (lv4i*)(unsigned)(uintptr_t)(lp + tid + 256), 0, 0);
                __builtin_amdgcn_global_load_async_to_lds_b128(
                    (gv4i*)(uintptr_t)(q4 + tid + 256),
                    (lv4i*)(unsigned)(uintptr_t)(lq + tid + 256), 0, 0);
            }
            asm volatile("s_wait_asynccnt 0" ::: "memory");
        }
#else
        // ---- fallback: stage d = p - q through VGPRs ----
        {
            float4* dd = (float4*)d_tile;
            float4 a = p4[tid], b = q4[tid];
            dd[tid] = make_float4(a.x - b.x, a.y - b.y, a.z - b.z, a.w - b.w);
            if (tid < TILE_F4 - 256) {
                float4 a2 = p4[tid + 256], b2 = q4[tid + 256];
                dd[tid + 256] = make_float4(a2.x - b2.x, a2.y - b2.y,
                                            a2.z - b2.z, a2.w - b2.w);
            }
        }
#endif
        __syncthreads();

        // ---- per-wave: cdf for 16 samples via triangular WMMA ----
#ifdef HAVE_ASYNC_LDS
        const float* psr = p_tile + (wave * 16 + n) * C_DIM;
        const float* qsr = q_tile + (wave * 16 + n) * C_DIM;
#define D_AT(k) (psr[(k)] - qsr[(k)])
#else
        const float* dsr = d_tile + (wave * 16 + n) * C_DIM;
#define D_AT(k) (dsr[(k)])
#endif

        float part = 0.0f;
#if __has_builtin(__builtin_amdgcn_wmma_f32_16x16x4_f32)
        v8f acc = {0.f, 0.f, 0.f, 0.f, 0.f, 0.f, 0.f, 0.f};
#pragma unroll
        for (int c = 0; c < 3; ++c) {          // K chunks: 0..3, 4..7, 8..11
            v2f av, bv;
#pragma unroll
            for (int v = 0; v < 2; ++v) {
                const int kg = 4 * c + v + 2 * hi;        // global K index
                const int kc = kg < C_DIM ? kg : 0;       // clamped LDS index
                const float lval = D_AT(kc);
                bv[v] = (kg < C_DIM) ? lval : 0.0f;       // B: d[kg][n], zero pad
                av[v] = (kg <= n && kg < C_DIM) ? 1.0f : 0.0f;  // A: triangular ones
            }
            acc = __builtin_amdgcn_wmma_f32_16x16x4_f32(
                false, av, false, bv, (short)0, acc, false, false);
        }
        // D rows: lanes 0-15 hold m = v (0..7); lanes 16-31 hold m = v+8 (8..15).
#pragma unroll
        for (int v = 0; v < 8; ++v) {
            const int mrow = v + 8 * hi;
            const float cv = (mrow < C_DIM) ? acc[v] : 0.0f;
            part += cv * cv;
        }
#else
        // Scalar fallback: each half-wave covers its row range of sample n.
        {
            const int mlo = hi ? 8 : 0;
            const int mhi = hi ? (C_DIM - 1) : 7;
            float run = 0.0f;
            for (int k = 0; k <= mhi; ++k) {
                run += D_AT(k);
                if (k >= mlo) part += run * run;
            }
        }
#endif
#undef D_AT
        // combine the two half-wave partials for sample n (lanes L and L^16)
        float total = part + __shfl_xor(part, 16, 32);
#if __has_builtin(__builtin_amdgcn_sqrtf)
        float persample = __builtin_amdgcn_sqrtf(total * (1.0f / (float)C_DIM));
#else
        float persample = sqrtf(total * (1.0f / (float)C_DIM));
#endif
        // sum the 16 distinct samples within each half-wave
        persample += __shfl_xor(persample, 8, 32);
        persample += __shfl_xor(persample, 4, 32);
        persample += __shfl_xor(persample, 2, 32);
        persample += __shfl_xor(persample, 1, 32);
        if (lane == 0) atomicAdd(&block_acc, persample);  // ds_add_f32
    }
    __syncthreads();
    if (tid == 0) atomicAdd(out, block_acc * invB);
}

// Tail for B % SPB != 0 (not hit for B = 2,000,000, kept for generality).
__global__ void emd_tail_kernel(const float* __restrict__ p,
                                const float* __restrict__ q,
                                float* __restrict__ out,
                                int start, int Btot, float invB) {
    int s = start + blockIdx.x * blockDim.x + threadIdx.x;
    if (s >= Btot) return;
    const float* ps = p + (size_t)s * C_DIM;
    const float* qs = q + (size_t)s * C_DIM;
    float run = 0.0f, acc = 0.0f;
#pragma unroll
    for (int k = 0; k < C_DIM; ++k) {
        run += ps[k] - qs[k];
        acc += run * run;
    }
    atomicAdd(out, sqrtf(acc * (1.0f / (float)C_DIM)) * invB);
}

extern "C" void kernel_launch(void* const* d_in, const int* in_sizes, int n_in,
                              void* d_out, int out_size, void* d_ws, size_t ws_size,
                              hipStream_t stream) {
    (void)n_in; (void)out_size; (void)d_ws; (void)ws_size;
    const float* p = (const float*)d_in[0];
    const float* q = (const float*)d_in[1];
    // d_in[2] is r (int, ==2 per setup); r=2 is compiled in.
    float* out = (float*)d_out;

    const long total = in_sizes[0];           // B * C
    const int  B     = (int)(total / C_DIM);
    const int  ntiles  = B / SPB;
    const int  covered = ntiles * SPB;
    const float invB = 1.0f / (float)B;

    emd_zero_out<<<1, 32, 0, stream>>>(out);

    if (ntiles > 0) {
        int grid = ntiles < 8192 ? ntiles : 8192;   // grid-stride over tiles
        emd_wmma_kernel<<<grid, 256, 0, stream>>>(p, q, out, ntiles, invB);
    }
    if (covered < B) {
        int rem = B - covered;
        emd_tail_kernel<<<(rem + 255) / 256, 256, 0, stream>>>(p, q, out,
                                                               covered, B, invB);
    }
}